// Extractor_N2V_56848187130529
// MI455X (gfx1250) — compile-verified
//
#include <hip/hip_runtime.h>

#define N_NODES 100000
#define E_EDGES 3200000
#define IN_DIM  128
#define HID     256
#define BN_EPS  1e-5f

typedef float v2f __attribute__((ext_vector_type(2)));
typedef float v8f __attribute__((ext_vector_type(8)));

// ---------------------------------------------------------------------------
// GEMM1: x[N,256] = h[N,128] @ W1[256,128]^T + b1
// One wave -> one 16x16 output tile via V_WMMA_F32_16X16X4_F32, K-loop step 4.
// Block = 256 threads = 8 waves covering 16 rows x 128 cols.
// ---------------------------------------------------------------------------
__global__ __launch_bounds__(256) void gemm1_wmma(const float* __restrict__ h,
                                                  const float* __restrict__ W1,
                                                  const float* __restrict__ b1,
                                                  float* __restrict__ x) {
  const int lane = threadIdx.x & 31;
  const int wave = threadIdx.x >> 5;
  const int rowTile = blockIdx.y * 16;
  const int colTile = blockIdx.x * 128 + wave * 16;
  const int n  = lane & 15;   // A row index / B-D column index within tile
  const int hi = lane >> 4;   // half-wave select (K+2 / M+8)

  const float* aptr = h  + (size_t)(rowTile + n) * IN_DIM + 2 * hi;
  const float* bptr = W1 + (size_t)(colTile + n) * IN_DIM + 2 * hi;

  v8f acc = {};
  #pragma unroll 8
  for (int k = 0; k < IN_DIM; k += 4) {
    v2f a = *(const v2f*)(aptr + k);
    v2f b = *(const v2f*)(bptr + k);
    acc = __builtin_amdgcn_wmma_f32_16x16x4_f32(false, a, false, b,
                                                (short)0, acc, false, false);
  }

  const float bias = b1[colTile + n];
  float* orow = x + (size_t)(rowTile + 8 * hi) * HID + colTile + n;
  #pragma unroll
  for (int r = 0; r < 8; ++r)
    orow[(size_t)r * HID] = acc[r] + bias;
}

// ---------------------------------------------------------------------------
// Edge scatter: pooled[dst] += x[src] (256 ch), deg[dst] += 1.
// One wave per edge; each lane moves 8 channels (two float4 loads, 8 atomics).
// ---------------------------------------------------------------------------
__global__ __launch_bounds__(256) void scatter_edges(const float* __restrict__ x,
                                                     const int* __restrict__ esrc,
                                                     const int* __restrict__ edst,
                                                     float* __restrict__ pooled,
                                                     float* __restrict__ deg) {
  const int lane = threadIdx.x & 31;
  const int wave = threadIdx.x >> 5;
  const long long e = (long long)blockIdx.x * 8 + wave;
  if (e >= E_EDGES) return;
  const int s = esrc[e];
  const int d = edst[e];

  const float4* xs = (const float4*)(x + (size_t)s * HID);
  float* pd = pooled + (size_t)d * HID;

  float4 v0 = xs[lane];        // channels 4*lane .. 4*lane+3
  float4 v1 = xs[lane + 32];   // channels 128 + 4*lane ..
  int c0 = lane * 4;
  atomicAdd(pd + c0 + 0, v0.x);
  atomicAdd(pd + c0 + 1, v0.y);
  atomicAdd(pd + c0 + 2, v0.z);
  atomicAdd(pd + c0 + 3, v0.w);
  atomicAdd(pd + 128 + c0 + 0, v1.x);
  atomicAdd(pd + 128 + c0 + 1, v1.y);
  atomicAdd(pd + 128 + c0 + 2, v1.z);
  atomicAdd(pd + 128 + c0 + 3, v1.w);
  if (lane == 0) atomicAdd(deg + d, 1.0f);
}

// ---------------------------------------------------------------------------
// Combine (pooled/deg + eps*x, in-place into x) + per-channel sum/sumsq.
// Thread t owns channel t; block covers ROWS_PER_BLOCK rows; one atomic per
// channel per block at the end.
// ---------------------------------------------------------------------------
#define ROWS_PER_BLOCK 128
__global__ __launch_bounds__(256) void combine_stats(float* __restrict__ x,
                                                     const float* __restrict__ pooled,
                                                     const float* __restrict__ deg,
                                                     const float* __restrict__ eps1,
                                                     float* __restrict__ sums,
                                                     float* __restrict__ sumsq) {
  const int c = threadIdx.x;
  const int r0 = blockIdx.x * ROWS_PER_BLOCK;
  int r1 = r0 + ROWS_PER_BLOCK;
  if (r1 > N_NODES) r1 = N_NODES;
  const float eps = eps1[0];

  float acc = 0.0f, acc2 = 0.0f;
  for (int r = r0; r < r1; ++r) {
    const float dg = deg[r];
    const size_t idx = (size_t)r * HID + c;
    const float v = pooled[idx] / dg + eps * x[idx];
    x[idx] = v;
    acc  += v;
    acc2 += v * v;
  }
  atomicAdd(&sums[c],  acc);
  atomicAdd(&sumsq[c], acc2);
}

// ---------------------------------------------------------------------------
// BN finalize: scale[c] = gamma*rsqrt(var+eps); shift[c] = beta - mean*scale.
// ---------------------------------------------------------------------------
__global__ __launch_bounds__(256) void finalize_stats(const float* __restrict__ sums,
                                                      const float* __restrict__ sumsq,
                                                      const float* __restrict__ gamma,
                                                      const float* __restrict__ beta,
                                                      float* __restrict__ scale,
                                                      float* __restrict__ shift) {
  const int c = threadIdx.x;
  const float inv_n = 1.0f / (float)N_NODES;
  const float mean = sums[c] * inv_n;
  const float var  = sumsq[c] * inv_n - mean * mean;
  const float s    = rsqrtf(var + BN_EPS) * gamma[c];
  scale[c] = s;
  shift[c] = beta[c] - mean * s;
}

// ---------------------------------------------------------------------------
// GEMM2: out[N,256] = BN(x)[N,256] @ W2[256,256]^T + b2, with the per-channel
// affine BN folded into the A-fragment load (K dim == BN channel dim).
// ---------------------------------------------------------------------------
__global__ __launch_bounds__(256) void gemm2_wmma(const float* __restrict__ x,
                                                  const float* __restrict__ W2,
                                                  const float* __restrict__ b2,
                                                  const float* __restrict__ scale,
                                                  const float* __restrict__ shift,
                                                  float* __restrict__ out) {
  __shared__ float s_scale[HID];
  __shared__ float s_shift[HID];
  s_scale[threadIdx.x] = scale[threadIdx.x];
  s_shift[threadIdx.x] = shift[threadIdx.x];
  __syncthreads();

  const int lane = threadIdx.x & 31;
  const int wave = threadIdx.x >> 5;
  const int rowTile = blockIdx.y * 16;
  const int colTile = blockIdx.x * 128 + wave * 16;
  const int n  = lane & 15;
  const int hi = lane >> 4;

  const float* aptr = x  + (size_t)(rowTile + n) * HID + 2 * hi;
  const float* bptr = W2 + (size_t)(colTile + n) * HID + 2 * hi;

  v8f acc = {};
  #pragma unroll 4
  for (int k = 0; k < HID; k += 4) {
    const int kb = k + 2 * hi;
    v2f a = *(const v2f*)(aptr + k);
    a.x = a.x * s_scale[kb]     + s_shift[kb];
    a.y = a.y * s_scale[kb + 1] + s_shift[kb + 1];
    v2f b = *(const v2f*)(bptr + k);
    acc = __builtin_amdgcn_wmma_f32_16x16x4_f32(false, a, false, b,
                                                (short)0, acc, false, false);
  }

  const float bias = b2[colTile + n];
  float* orow = out + (size_t)(rowTile + 8 * hi) * HID + colTile + n;
  #pragma unroll
  for (int r = 0; r < 8; ++r)
    orow[(size_t)r * HID] = acc[r] + bias;
}

// ---------------------------------------------------------------------------
extern "C" void kernel_launch(void* const* d_in, const int* in_sizes, int n_in,
                              void* d_out, int out_size, void* d_ws, size_t ws_size,
                              hipStream_t stream) {
  (void)in_sizes; (void)n_in; (void)out_size; (void)ws_size;

  const float* h     = (const float*)d_in[0];
  const int*   esrc  = (const int*)  d_in[1];
  const int*   edst  = (const int*)  d_in[2];
  const float* W1    = (const float*)d_in[3];
  const float* b1    = (const float*)d_in[4];
  const float* W2    = (const float*)d_in[5];
  const float* b2    = (const float*)d_in[6];
  const float* gamma = (const float*)d_in[7];
  const float* beta  = (const float*)d_in[8];
  const float* eps1  = (const float*)d_in[9];
  float* out = (float*)d_out;

  // Workspace layout (fp32): x | pooled | deg | sums | sumsq | scale | shift
  float* ws     = (float*)d_ws;
  float* xbuf   = ws;
  float* pooled = xbuf   + (size_t)N_NODES * HID;
  float* deg    = pooled + (size_t)N_NODES * HID;
  float* sums   = deg    + N_NODES;
  float* sumsq  = sums   + HID;
  float* scale  = sumsq  + HID;
  float* shift  = scale  + HID;

  // Zero the accumulation region (pooled..sumsq is contiguous).
  hipMemsetAsync(pooled, 0,
                 ((size_t)N_NODES * HID + N_NODES + 2 * HID) * sizeof(float),
                 stream);

  // GEMM1: 16-row tiles x 128-col per block.
  dim3 g1(HID / 128, N_NODES / 16);
  gemm1_wmma<<<g1, 256, 0, stream>>>(h, W1, b1, xbuf);

  // Edge scatter: 8 edges (8 waves) per block.
  scatter_edges<<<(E_EDGES + 7) / 8, 256, 0, stream>>>(xbuf, esrc, edst, pooled, deg);

  // Combine + BN statistics.
  combine_stats<<<(N_NODES + ROWS_PER_BLOCK - 1) / ROWS_PER_BLOCK, 256, 0, stream>>>(
      xbuf, pooled, deg, eps1, sums, sumsq);

  finalize_stats<<<1, 256, 0, stream>>>(sums, sumsq, gamma, beta, scale, shift);

  // GEMM2 with BN folded into A loads.
  dim3 g2(HID / 128, N_NODES / 16);
  gemm2_wmma<<<g2, 256, 0, stream>>>(xbuf, W2, b2, scale, shift, out);
}